// SlotAttention_35837207117944
// MI455X (gfx1250) — compile-verified
//
#include <hip/hip_runtime.h>
#include <hip/hip_bf16.h>

typedef float v2f __attribute__((ext_vector_type(2)));
typedef float v8f __attribute__((ext_vector_type(8)));

#define BB 64
#define NN 4096
#define DD 128
#define SS 7
#define HID 256

// ---------------------------------------------------------------------------
// Kernel 1: LayerNorm(x) then k = xn @ Wk^T, v = xn @ Wv^T via f32 WMMA.
// One wave handles a 16-row tile; 4 waves per block.
// ---------------------------------------------------------------------------
__global__ __launch_bounds__(128) void ln_kv_kernel(
    const float* __restrict__ x, const float* __restrict__ Wk,
    const float* __restrict__ Wv, const float* __restrict__ g,
    const float* __restrict__ bt, float* __restrict__ kout,
    float* __restrict__ vout) {
  __shared__ float xs[4][16][132];  // padded stride to dodge bank conflicts
  const int tid = threadIdx.x;
  const int wave = tid >> 5, lane = tid & 31;
  const long row0 = ((long)blockIdx.x * 4 + wave) * 16;

  // Stage 16x128 tile of x into LDS (float4 per lane per row).
  for (int r = 0; r < 16; ++r) {
    float4 val = *(const float4*)(x + (row0 + r) * DD + lane * 4);
    *(float4*)&xs[wave][r][lane * 4] = val;
  }
  __syncthreads();

  // LayerNorm: lanes 0..15 each own one row.
  if (lane < 16) {
    float s = 0.f, s2 = 0.f;
    for (int c = 0; c < DD; ++c) { float v = xs[wave][lane][c]; s += v; s2 += v * v; }
    float m = s * (1.f / DD);
    float var = s2 * (1.f / DD) - m * m;
    float rstd = rsqrtf(var + 1e-5f);
    for (int c = 0; c < DD; ++c) {
      float v = xs[wave][lane][c];
      xs[wave][lane][c] = (v - m) * rstd * g[c] + bt[c];
    }
  }
  __syncthreads();

  const int rowa = lane & 15;
  const int koff = (lane >> 4) * 2;   // A/B fragment K-offset per half-wave
  const int rowoff = (lane >> 4) * 8; // D fragment row offset per half-wave
  const float* arow = &xs[wave][rowa][koff];

  for (int nt = 0; nt < 8; ++nt) {
    const int colb = nt * 16 + (lane & 15);
    const float* wkrow = Wk + colb * DD + koff;
    const float* wvrow = Wv + colb * DD + koff;
    v8f acck{}; v8f accv{};
    for (int kk = 0; kk < 32; ++kk) {
      v2f a  = *(const v2f*)(arow + kk * 4);
      v2f bk = *(const v2f*)(wkrow + kk * 4);
      v2f bv = *(const v2f*)(wvrow + kk * 4);
      acck = __builtin_amdgcn_wmma_f32_16x16x4_f32(false, a, false, bk, (short)0, acck, false, false);
      accv = __builtin_amdgcn_wmma_f32_16x16x4_f32(false, a, false, bv, (short)0, accv, false, false);
    }
    for (int m = 0; m < 8; ++m) {
      kout[(row0 + rowoff + m) * DD + colb] = acck[m];
      vout[(row0 + rowoff + m) * DD + colb] = accv[m];
    }
  }
}

// ---------------------------------------------------------------------------
// Kernel 2: slots = mu + (softplus(logsigma)+1e-5) * noise
// ---------------------------------------------------------------------------
__global__ void slots_init_kernel(const float* __restrict__ noise,
                                  const float* __restrict__ mu,
                                  const float* __restrict__ logsigma,
                                  float* __restrict__ slots, int n) {
  int i = blockIdx.x * blockDim.x + threadIdx.x;
  if (i >= n) return;
  int d = i & (DD - 1);
  float ls = logsigma[d];
  float sp = (ls > 20.f) ? ls : log1pf(__expf(ls));
  slots[i] = mu[d] + (sp + 1e-5f) * noise[i];
}

// ---------------------------------------------------------------------------
// Kernel 3 (per iter): sn = LN(slots); q = sn @ Wq^T. Also zero sumN/updates.
// One block per (b,s) row, 128 threads.
// ---------------------------------------------------------------------------
__global__ __launch_bounds__(128) void qprep_kernel(
    const float* __restrict__ slots, const float* __restrict__ Wq,
    const float* __restrict__ g, const float* __restrict__ bt,
    float* __restrict__ q, float* __restrict__ sumN,
    float* __restrict__ updates) {
  __shared__ float sn[DD], red[DD], red2[DD];
  const int row = blockIdx.x, t = threadIdx.x;
  float v = slots[row * DD + t];
  red[t] = v; red2[t] = v * v;
  __syncthreads();
  for (int s = 64; s > 0; s >>= 1) {
    if (t < s) { red[t] += red[t + s]; red2[t] += red2[t + s]; }
    __syncthreads();
  }
  float m = red[0] * (1.f / DD);
  float var = red2[0] * (1.f / DD) - m * m;
  float rstd = rsqrtf(var + 1e-5f);
  sn[t] = (v - m) * rstd * g[t] + bt[t];
  updates[row * DD + t] = 0.f;
  if (t == 0) sumN[row] = 0.f;
  __syncthreads();
  const float* w = Wq + t * DD;
  float acc = 0.f;
  for (int c = 0; c < DD; c += 4) {
    float4 w4 = *(const float4*)(w + c);
    acc += w4.x * sn[c] + w4.y * sn[c + 1] + w4.z * sn[c + 2] + w4.w * sn[c + 3];
  }
  q[row * DD + t] = acc;
}

// ---------------------------------------------------------------------------
// Kernel 4 (per iter): scores = k @ q^T / sqrt(D) via WMMA, softmax over S,
// +EPS, store attn, accumulate column sums over N into sumN.
// grid (B, N/64), 4 waves x 16 rows each.
// ---------------------------------------------------------------------------
__global__ __launch_bounds__(128) void scores_kernel(
    const float* __restrict__ k, const float* __restrict__ q,
    float* __restrict__ attn, float* __restrict__ sumN) {
  __shared__ float qs[16 * 132];     // q^T tile, rows 7..15 zero
  __shared__ float sc[4][16 * 17];   // per-wave score scratch
  const int b = blockIdx.x;
  const int tid = threadIdx.x, wave = tid >> 5, lane = tid & 31;
  for (int i = tid; i < 16 * 132; i += 128) qs[i] = 0.f;
  __syncthreads();
  for (int i = tid; i < SS * DD; i += 128) {
    int s = i >> 7, c = i & (DD - 1);
    qs[s * 132 + c] = q[(b * SS + s) * DD + c];
  }
  __syncthreads();

  const long n0 = (long)blockIdx.y * 64 + wave * 16;
  const int rowa = lane & 15, koff = (lane >> 4) * 2, rowoff = (lane >> 4) * 8;
  const float* krow = k + ((long)b * NN + n0 + rowa) * DD + koff;
  const float* qrow = &qs[(lane & 15) * 132 + koff];
  v8f acc{};
  for (int kk = 0; kk < 32; ++kk) {
    v2f a  = *(const v2f*)(krow + kk * 4);
    v2f bq = *(const v2f*)(qrow + kk * 4);
    acc = __builtin_amdgcn_wmma_f32_16x16x4_f32(false, a, false, bq, (short)0, acc, false, false);
  }
  const float scale = 0.088388347648318447f; // 1/sqrt(128)
  for (int m = 0; m < 8; ++m)
    sc[wave][(rowoff + m) * 17 + (lane & 15)] = acc[m] * scale;
  __syncthreads();

  float p[SS];
  if (lane < 16) {
    const float* srow = &sc[wave][lane * 17];
    float mx = srow[0];
    for (int s = 1; s < SS; ++s) mx = fmaxf(mx, srow[s]);
    float sum = 0.f;
    for (int s = 0; s < SS; ++s) { p[s] = __expf(srow[s] - mx); sum += p[s]; }
    float inv = 1.f / sum;
    for (int s = 0; s < SS; ++s) {
      p[s] = p[s] * inv + 1e-8f;
      attn[((long)b * NN + n0 + lane) * SS + s] = p[s];
    }
  } else {
    for (int s = 0; s < SS; ++s) p[s] = 0.f;
  }
  // Wave-wide reduction of per-slot partial sums, then one atomic per slot.
  for (int s = 0; s < SS; ++s) {
    float tt = p[s];
    for (int msk = 1; msk < 32; msk <<= 1) tt += __shfl_xor(tt, msk, 32);
    if (lane == 0) atomicAdd(&sumN[b * SS + s], tt);
  }
}

// ---------------------------------------------------------------------------
// Kernel 5 (per iter): updates[b,s,:] += sum_n attn[b,n,s] * v[b,n,:]
// grid (B, N/256); thread t owns dimension d=t; attn chunk staged in LDS.
// ---------------------------------------------------------------------------
__global__ __launch_bounds__(128) void updacc_kernel(
    const float* __restrict__ v, const float* __restrict__ attn,
    float* __restrict__ updates) {
  __shared__ float at[256 * SS];
  const int b = blockIdx.x, t = threadIdx.x;
  const long n0 = (long)blockIdx.y * 256;
  for (int i = t; i < 256 * SS; i += 128) at[i] = attn[((long)b * NN + n0) * SS + i];
  __syncthreads();
  float acc[SS] = {0.f, 0.f, 0.f, 0.f, 0.f, 0.f, 0.f};
  const float* vp = v + ((long)b * NN + n0) * DD + t;
  for (int n = 0; n < 256; ++n) {
    float vv = vp[n * DD];
#pragma unroll
    for (int s = 0; s < SS; ++s) acc[s] = fmaf(at[n * SS + s], vv, acc[s]);
  }
  for (int s = 0; s < SS; ++s) atomicAdd(&updates[(b * SS + s) * DD + t], acc[s]);
}

// ---------------------------------------------------------------------------
// Kernel 6 (per iter): renorm updates, GRU cell, LayerNorm, MLP residual.
// One block per (b,s) row, 128 threads.
// ---------------------------------------------------------------------------
__global__ __launch_bounds__(128) void slotup_kernel(
    const float* __restrict__ updates, const float* __restrict__ sumN,
    const float* __restrict__ slots,
    const float* __restrict__ W_ih, const float* __restrict__ W_hh,
    const float* __restrict__ b_ih, const float* __restrict__ b_hh,
    const float* __restrict__ g, const float* __restrict__ bt,
    const float* __restrict__ W1, const float* __restrict__ b1,
    const float* __restrict__ W2, const float* __restrict__ b2,
    float* __restrict__ out) {
  __shared__ float u[DD], h[DD], red[DD], red2[DD], sln[DD], hid[HID];
  const int row = blockIdx.x, t = threadIdx.x;
  float inv = 1.f / sumN[row];
  u[t] = updates[row * DD + t] * inv;
  float hv = slots[row * DD + t];
  h[t] = hv;
  __syncthreads();
  float gx[3], gh[3];
  for (int gi = 0; gi < 3; ++gi) {
    const float* wi = W_ih + (gi * DD + t) * DD;
    const float* wh = W_hh + (gi * DD + t) * DD;
    float ax = 0.f, ah = 0.f;
    for (int c = 0; c < DD; c += 4) {
      float4 a4 = *(const float4*)(wi + c);
      float4 b4 = *(const float4*)(wh + c);
      ax += a4.x * u[c] + a4.y * u[c + 1] + a4.z * u[c + 2] + a4.w * u[c + 3];
      ah += b4.x * h[c] + b4.y * h[c + 1] + b4.z * h[c + 2] + b4.w * h[c + 3];
    }
    gx[gi] = ax + b_ih[gi * DD + t];
    gh[gi] = ah + b_hh[gi * DD + t];
  }
  float r = 1.f / (1.f + __expf(-(gx[0] + gh[0])));
  float z = 1.f / (1.f + __expf(-(gx[1] + gh[1])));
  float nn = tanhf(gx[2] + r * gh[2]);
  float hn = (1.f - z) * nn + z * hv;
  // LayerNorm(ln_mlp)
  red[t] = hn; red2[t] = hn * hn;
  __syncthreads();
  for (int s = 64; s > 0; s >>= 1) {
    if (t < s) { red[t] += red[t + s]; red2[t] += red2[t + s]; }
    __syncthreads();
  }
  float m = red[0] * (1.f / DD);
  float var = red2[0] * (1.f / DD) - m * m;
  float rstd = rsqrtf(var + 1e-5f);
  float sv = (hn - m) * rstd * g[t] + bt[t];
  sln[t] = sv;
  __syncthreads();
  for (int j = t; j < HID; j += 128) {
    const float* w = W1 + j * DD;
    float a = 0.f;
    for (int c = 0; c < DD; c += 4) {
      float4 w4 = *(const float4*)(w + c);
      a += w4.x * sln[c] + w4.y * sln[c + 1] + w4.z * sln[c + 2] + w4.w * sln[c + 3];
    }
    hid[j] = fmaxf(a + b1[j], 0.f);
  }
  __syncthreads();
  const float* w = W2 + t * HID;
  float a = 0.f;
  for (int c = 0; c < HID; c += 4) {
    float4 w4 = *(const float4*)(w + c);
    a += w4.x * hid[c] + w4.y * hid[c + 1] + w4.z * hid[c + 2] + w4.w * hid[c + 3];
  }
  out[row * DD + t] = sv + a + b2[t];
}

// ---------------------------------------------------------------------------
extern "C" void kernel_launch(void* const* d_in, const int* in_sizes, int n_in,
                              void* d_out, int out_size, void* d_ws, size_t ws_size,
                              hipStream_t stream) {
  const float* x        = (const float*)d_in[0];
  const float* noise    = (const float*)d_in[1];
  const float* Wq       = (const float*)d_in[2];
  const float* Wk       = (const float*)d_in[3];
  const float* Wv       = (const float*)d_in[4];
  const float* ln_in_g  = (const float*)d_in[5];
  const float* ln_in_b  = (const float*)d_in[6];
  const float* ln_sl_g  = (const float*)d_in[7];
  const float* ln_sl_b  = (const float*)d_in[8];
  const float* ln_mlp_g = (const float*)d_in[9];
  const float* ln_mlp_b = (const float*)d_in[10];
  const float* W_ih     = (const float*)d_in[11];
  const float* W_hh     = (const float*)d_in[12];
  const float* b_ih     = (const float*)d_in[13];
  const float* b_hh     = (const float*)d_in[14];
  const float* W1       = (const float*)d_in[15];
  const float* b1       = (const float*)d_in[16];
  const float* W2       = (const float*)d_in[17];
  const float* b2       = (const float*)d_in[18];
  const float* mu       = (const float*)d_in[19];
  const float* logsigma = (const float*)d_in[20];
  (void)in_sizes; (void)n_in; (void)ws_size; (void)out_size;

  char* ws = (char*)d_ws;
  const size_t KV_BYTES   = (size_t)BB * NN * DD * sizeof(float); // 128 MB
  const size_t ATTN_BYTES = (size_t)BB * NN * SS * sizeof(float);
  const size_t ROW_BYTES  = (size_t)BB * SS * DD * sizeof(float);
  float* k       = (float*)(ws);
  float* v       = (float*)(ws + KV_BYTES);
  float* attn    = (float*)(ws + 2 * KV_BYTES);
  float* slots   = (float*)(ws + 2 * KV_BYTES + ATTN_BYTES);
  float* q       = (float*)(ws + 2 * KV_BYTES + ATTN_BYTES + ROW_BYTES);
  float* updates = (float*)(ws + 2 * KV_BYTES + ATTN_BYTES + 2 * ROW_BYTES);
  float* sumN    = (float*)(ws + 2 * KV_BYTES + ATTN_BYTES + 3 * ROW_BYTES);

  // 1) LN + k/v projections (WMMA f32).
  ln_kv_kernel<<<(BB * NN) / (16 * 4), 128, 0, stream>>>(x, Wk, Wv, ln_in_g,
                                                         ln_in_b, k, v);
  // 2) slots rsample.
  const int nslot = BB * SS * DD;
  slots_init_kernel<<<(nslot + 255) / 256, 256, 0, stream>>>(noise, mu,
                                                             logsigma, slots, nslot);
  // 3) iterations
  for (int it = 0; it < 3; ++it) {
    qprep_kernel<<<BB * SS, 128, 0, stream>>>(slots, Wq, ln_sl_g, ln_sl_b, q,
                                              sumN, updates);
    scores_kernel<<<dim3(BB, NN / 64), 128, 0, stream>>>(k, q, attn, sumN);
    updacc_kernel<<<dim3(BB, NN / 256), 128, 0, stream>>>(v, attn, updates);
    float* outp = (it == 2) ? (float*)d_out : slots;
    slotup_kernel<<<BB * SS, 128, 0, stream>>>(updates, sumN, slots, W_ih, W_hh,
                                               b_ih, b_hh, ln_mlp_g, ln_mlp_b,
                                               W1, b1, W2, b2, outp);
  }
}